// GraphSAGE_91199335563655
// MI455X (gfx1250) — compile-verified
//
#include <hip/hip_runtime.h>

#define D 128
#define KFAN 20
#define ENTRIES_PER_BLOCK 256

typedef __attribute__((ext_vector_type(2))) float v2f;
typedef __attribute__((ext_vector_type(8))) float v8f;
typedef int v4i __attribute__((vector_size(16)));

#if defined(__AMDGCN__) && __has_builtin(__builtin_amdgcn_global_load_async_to_lds_b128)
#define HAS_ASYNC_LDS 1
#else
#define HAS_ASYNC_LDS 0
#endif

// Copy one 512B row (128 floats) global -> LDS, one B128 per lane of a wave.
__device__ __forceinline__ void copy_row_to_lds(const float* __restrict__ src,
                                                float* dstLds, int lane) {
#if HAS_ASYNC_LDS
  __builtin_amdgcn_global_load_async_to_lds_b128(
      (v4i*)(src + 4 * lane), (v4i*)(dstLds + 4 * lane), 0, 0);
#else
  float4 v = *(const float4*)(src + 4 * lane);
  *(float4*)(dstLds + 4 * lane) = v;
#endif
}

__device__ __forceinline__ void wait_lds_fill() {
#if HAS_ASYNC_LDS
#if __has_builtin(__builtin_amdgcn_s_wait_asynccnt)
  __builtin_amdgcn_s_wait_asynccnt(0);
#else
  asm volatile("s_wait_asynccnt 0" ::: "memory");
#endif
#endif
  __syncthreads();
}

// ---------------------------------------------------------------------------
// Segment-sum for sorted ragged bags. One wave per block; lane owns 4 dims
// (b128 table reads). Run-length accumulate in registers; atomics only at
// segment boundaries (avg run ~10 -> ~10x fewer atomics).
// ---------------------------------------------------------------------------
__global__ void bag_scatter_kernel(const float* __restrict__ table,
                                   const int* __restrict__ idx,
                                   const int* __restrict__ bag, int T,
                                   float* __restrict__ sum,
                                   float* __restrict__ cnt) {
  const int lane = threadIdx.x;  // 0..31
  int t0 = blockIdx.x * ENTRIES_PER_BLOCK;
  if (t0 >= T) return;
  int t1 = min(t0 + ENTRIES_PER_BLOCK, T);
  int cur = bag[t0];
  float ax = 0.f, ay = 0.f, az = 0.f, aw = 0.f;
  int run = 0;
  for (int t = t0; t < t1; ++t) {
    const int b = bag[t];
    if (b != cur) {
      float* s = &sum[(size_t)cur * D + 4 * lane];
      atomicAdd(s + 0, ax); atomicAdd(s + 1, ay);
      atomicAdd(s + 2, az); atomicAdd(s + 3, aw);
      if (lane == 0) atomicAdd(&cnt[cur], (float)run);
      cur = b; ax = ay = az = aw = 0.f; run = 0;
    }
    const float4 v = *(const float4*)(table + (size_t)idx[t] * D + 4 * lane);
    ax += v.x; ay += v.y; az += v.z; aw += v.w;
    ++run;
  }
  float* s = &sum[(size_t)cur * D + 4 * lane];
  atomicAdd(s + 0, ax); atomicAdd(s + 1, ay);
  atomicAdd(s + 2, az); atomicAdd(s + 3, aw);
  if (lane == 0) atomicAdd(&cnt[cur], (float)run);
}

// ---------------------------------------------------------------------------
// cnt[i] -> 1/max(cnt[i],1): one IEEE divide per node (150K total) instead of
// one per output element (19.2M) in the WMMA epilogue.
// ---------------------------------------------------------------------------
__global__ void rcp_cnt_kernel(float* __restrict__ cnt, int N) {
  const int i = blockIdx.x * blockDim.x + threadIdx.x;
  if (i < N) cnt[i] = 1.0f / fmaxf(cnt[i], 1.0f);
}

// ---------------------------------------------------------------------------
// Wave-level fp32 WMMA tile: one 16x16 output tile per wave via chained
// V_WMMA_F32_16X16X4_F32. A from padded LDS tile, B (KD x 128 row-major)
// from global (L2-resident weights). Returns raw accumulator.
// ISA 7.12.2 layouts: A lanes 0-15: M=l,K+{0,1}; lanes 16-31: M=l-16,K+{2,3};
// B symmetric over N; C vgpr v = row v (+8 for hi half).
// ---------------------------------------------------------------------------
template <int KD, int LDA>
__device__ __forceinline__ v8f wave_wmma_acc(const float* __restrict__ Alds,
                                             const float* __restrict__ Wg,
                                             int cbase, int half, int l16) {
  v8f c = {};
#pragma unroll 4
  for (int k = 0; k < KD; k += 4) {
    const int ka = k + half * 2;
    v2f a, b;
    a.x = Alds[l16 * LDA + ka];
    a.y = Alds[l16 * LDA + ka + 1];
    b.x = Wg[(size_t)ka * D + cbase + l16];
    b.y = Wg[(size_t)(ka + 1) * D + cbase + l16];
    c = __builtin_amdgcn_wmma_f32_16x16x4_f32(false, a, false, b, (short)0, c,
                                              false, false);
  }
  return c;
}

// ---------------------------------------------------------------------------
// Finalize bag-mean + project, in place: out[m,:] = (sum[m,:] @ W)*rcp_c + b
// (row scale commutes with the GEMM, so the raw-sum tile is async-copied;
// rcp_cnt holds precomputed 1/max(count,1)).
// blockDim 256 (8 waves), 16 rows/block, grid exact (M % 16 == 0).
// ---------------------------------------------------------------------------
__global__ __launch_bounds__(256) void project_init_kernel(
    float* __restrict__ acc, const float* __restrict__ rcp_cnt,
    const float* __restrict__ Wg, const float* __restrict__ bias) {
  __shared__ float Atile[16 * (D + 4)];
  const size_t m0 = (size_t)blockIdx.x * 16;
  const int lane = threadIdx.x & 31;
  const int wave = threadIdx.x >> 5;
#pragma unroll
  for (int rr = 0; rr < 2; ++rr) {
    const int r = wave + rr * 8;
    copy_row_to_lds(acc + (m0 + r) * D, &Atile[r * (D + 4)], lane);
  }
  wait_lds_fill();

  const int cbase = wave * 16, half = lane >> 4, l16 = lane & 15;
  v8f c = wave_wmma_acc<D, D + 4>(Atile, Wg, cbase, half, l16);
  const float bv = bias[cbase + l16];
#pragma unroll
  for (int v = 0; v < 8; ++v) {
    const int row = v + half * 8;
    acc[(m0 + row) * D + cbase + l16] = c[v] * rcp_cnt[m0 + row] + bv;
  }
}

// ---------------------------------------------------------------------------
// GraphSAGE layer: out = act(concat(left[gather], right) @ W + b).
// leftIdx == nullptr -> identity rows for the left half.
// ---------------------------------------------------------------------------
template <bool RELU>
__global__ __launch_bounds__(256) void layer_gemm_kernel(
    const float* __restrict__ leftTab, const int* __restrict__ leftIdx,
    const float* __restrict__ right, const float* __restrict__ Wg,
    const float* __restrict__ bias, float* __restrict__ outp) {
  __shared__ float Atile[16 * (2 * D + 4)];
  const size_t m0 = (size_t)blockIdx.x * 16;
  const int lane = threadIdx.x & 31;
  const int wave = threadIdx.x >> 5;
#pragma unroll
  for (int rr = 0; rr < 2; ++rr) {
    const int r = wave + rr * 8;
    const size_t m = m0 + r;
    const size_t lrow = leftIdx ? (size_t)leftIdx[m] : m;
    copy_row_to_lds(leftTab + lrow * D, &Atile[r * (2 * D + 4)], lane);
    copy_row_to_lds(right + m * D, &Atile[r * (2 * D + 4) + D], lane);
  }
  wait_lds_fill();

  const int cbase = wave * 16, half = lane >> 4, l16 = lane & 15;
  v8f c = wave_wmma_acc<2 * D, 2 * D + 4>(Atile, Wg, cbase, half, l16);
  const float bv = bias[cbase + l16];
#pragma unroll
  for (int v = 0; v < 8; ++v) {
    float r = c[v] + bv;
    if (RELU) r = fmaxf(r, 0.0f);
    const int row = v + half * 8;
    outp[(m0 + row) * D + cbase + l16] = r;
  }
}

// ---------------------------------------------------------------------------
// Fan-out means (K=20). Wave per parent, lane owns 4 dims -> b128 loads.
// Gather on user_init (51 MB) mostly hits the 192 MB L2.
// ---------------------------------------------------------------------------
__global__ void agg_mean_gather_kernel(const float* __restrict__ src,
                                       const int* __restrict__ nbr,
                                       float* __restrict__ dst, int P) {
  const int lane = threadIdx.x & 31;
  const size_t p = (size_t)blockIdx.x * 4 + (threadIdx.x >> 5);
  if (p >= (size_t)P) return;
  float sx = 0.f, sy = 0.f, sz = 0.f, sw = 0.f;
  const int* nb = nbr + p * KFAN;
#pragma unroll
  for (int k = 0; k < KFAN; ++k) {
    const float4 v = *(const float4*)(src + (size_t)nb[k] * D + 4 * lane);
    sx += v.x; sy += v.y; sz += v.z; sw += v.w;
  }
  float4 o;
  o.x = sx * (1.0f / KFAN); o.y = sy * (1.0f / KFAN);
  o.z = sz * (1.0f / KFAN); o.w = sw * (1.0f / KFAN);
  *(float4*)(dst + p * D + 4 * lane) = o;
}

__global__ void agg_mean_seq_kernel(const float* __restrict__ src,
                                    float* __restrict__ dst, int P) {
  const int lane = threadIdx.x & 31;
  const size_t p = (size_t)blockIdx.x * 4 + (threadIdx.x >> 5);
  if (p >= (size_t)P) return;
  float sx = 0.f, sy = 0.f, sz = 0.f, sw = 0.f;
#pragma unroll
  for (int k = 0; k < KFAN; ++k) {
    const float4 v = *(const float4*)(src + (p * KFAN + k) * D + 4 * lane);
    sx += v.x; sy += v.y; sz += v.z; sw += v.w;
  }
  float4 o;
  o.x = sx * (1.0f / KFAN); o.y = sy * (1.0f / KFAN);
  o.z = sz * (1.0f / KFAN); o.w = sw * (1.0f / KFAN);
  *(float4*)(dst + p * D + 4 * lane) = o;
}

// ---------------------------------------------------------------------------
extern "C" void kernel_launch(void* const* d_in, const int* in_sizes, int n_in,
                              void* d_out, int out_size, void* d_ws,
                              size_t ws_size, hipStream_t stream) {
  const float* user_feat_emb = (const float*)d_in[0];
  const float* item_feat_emb = (const float*)d_in[1];
  const float* Wu = (const float*)d_in[2];
  const float* bu = (const float*)d_in[3];
  const float* Wi = (const float*)d_in[4];
  const float* bi = (const float*)d_in[5];
  const float* W0 = (const float*)d_in[6];
  const float* b0 = (const float*)d_in[7];
  const float* W1 = (const float*)d_in[8];
  const float* b1 = (const float*)d_in[9];
  const int* user_feat_idx = (const int*)d_in[10];
  const int* user_bag_ids  = (const int*)d_in[11];
  const int* item_feat_idx = (const int*)d_in[12];
  const int* item_bag_ids  = (const int*)d_in[13];
  const int* n0 = (const int*)d_in[14];
  const int* n1 = (const int*)d_in[15];
  const int* n2 = (const int*)d_in[16];

  const int Tu = in_sizes[10];  // 1,000,000
  const int Ti = in_sizes[12];  //   500,000
  const int Bn = in_sizes[14];  // 2048
  const int BK = in_sizes[15];  // 40960
  const int NU = 100000, NI = 50000;

  // Workspace layout (floats). Accumulators first so one memset clears them.
  float* p = (float*)d_ws;
  float* user_acc = p; p += (size_t)NU * D;  // becomes user_init in place
  float* user_cnt = p; p += NU;
  float* item_acc = p; p += (size_t)NI * D;  // becomes item_init in place
  float* item_cnt = p; p += NI;
  const size_t zero_bytes = (size_t)((char*)p - (char*)d_ws);
  float* aggH2 = p; p += (size_t)BK * D;  // mean of h2 per n1 node
  float* aggH1 = p; p += (size_t)Bn * D;  // mean of h1 per n0 node
  float* g1    = p; p += (size_t)BK * D;
  float* g0    = p; p += (size_t)Bn * D;
  float* aggG1 = p; p += (size_t)Bn * D;

  (void)hipMemsetAsync(d_ws, 0, zero_bytes, stream);

  // 1) ragged bag sums (sorted segment scatter)
  bag_scatter_kernel<<<(Tu + ENTRIES_PER_BLOCK - 1) / ENTRIES_PER_BLOCK, 32, 0,
                       stream>>>(user_feat_emb, user_feat_idx, user_bag_ids,
                                 Tu, user_acc, user_cnt);
  bag_scatter_kernel<<<(Ti + ENTRIES_PER_BLOCK - 1) / ENTRIES_PER_BLOCK, 32, 0,
                       stream>>>(item_feat_emb, item_feat_idx, item_bag_ids,
                                 Ti, item_acc, item_cnt);

  // 1b) counts -> reciprocals (one divide per node)
  rcp_cnt_kernel<<<(NU + 255) / 256, 256, 0, stream>>>(user_cnt, NU);
  rcp_cnt_kernel<<<(NI + 255) / 256, 256, 0, stream>>>(item_cnt, NI);

  // 2) mean + projection (WMMA), in place -> node init tables
  project_init_kernel<<<NU / 16, 256, 0, stream>>>(user_acc, user_cnt, Wu, bu);
  project_init_kernel<<<NI / 16, 256, 0, stream>>>(item_acc, item_cnt, Wi, bi);

  // 3) neighbor means
  agg_mean_gather_kernel<<<(BK + 3) / 4, 128, 0, stream>>>(user_acc, n2,
                                                           aggH2, BK);
  agg_mean_gather_kernel<<<(Bn + 3) / 4, 128, 0, stream>>>(item_acc, n1,
                                                           aggH1, Bn);

  // 4) layer 0 (ReLU) for both depths (WMMA)
  layer_gemm_kernel<true><<<BK / 16, 256, 0, stream>>>(item_acc, n1, aggH2,
                                                       W0, b0, g1);
  layer_gemm_kernel<true><<<Bn / 16, 256, 0, stream>>>(user_acc, n0, aggH1,
                                                       W0, b0, g0);

  // 5) aggregate g1 and final layer (no ReLU) -> d_out
  agg_mean_seq_kernel<<<(Bn + 3) / 4, 128, 0, stream>>>(g1, aggG1, Bn);
  layer_gemm_kernel<false><<<Bn / 16, 256, 0, stream>>>(g0, nullptr, aggG1,
                                                        W1, b1, (float*)d_out);
}